// region_int_68264210202706
// MI455X (gfx1250) — compile-verified
//
#include <hip/hip_runtime.h>
#include <math.h>

#define D_R   333
#define D_IN  333
#define BATCH 4096

typedef float v2f __attribute__((ext_vector_type(2)));
typedef float v8f __attribute__((ext_vector_type(8)));

// ---------------------------------------------------------------------------
// Kernel 1: r_att[r] = sigmoid(r_id[r,:] @ att_w) ; normalize by (sum + 1e-8)
// ---------------------------------------------------------------------------
__global__ void att_kernel(const float* __restrict__ r_id,
                           const float* __restrict__ att_w,
                           float* __restrict__ att_ws,   // workspace copy (read by K3)
                           float* __restrict__ att_out)  // output slot
{
    __shared__ float red[512];
    const int r = threadIdx.x;
    float t = 0.f;
    if (r < D_R) {
        const float* row = r_id + (size_t)r * D_R;
        float acc = 0.f;
        for (int j = 0; j < D_R; ++j) acc += row[j] * att_w[j];
        t = 1.f / (1.f + expf(-acc));
    }
    red[threadIdx.x] = t;
    __syncthreads();
    for (int s = 256; s > 0; s >>= 1) {
        if (threadIdx.x < s) red[threadIdx.x] += red[threadIdx.x + s];
        __syncthreads();
    }
    const float total = red[0] + 1e-8f;
    if (r < D_R) {
        const float v = t / total;
        att_ws[r]  = v;
        att_out[r] = v;
    }
}

// ---------------------------------------------------------------------------
// Kernel 2: r_out[b,r] = x[b,r,:] @ W[r,:] + bias[r]  via V_WMMA_F32_16X16X4_F32
//
// One wave per (16-batch tile, region r).
//   A (16x4, MxK):  lane-half h, vgpr j -> x[b0 + (lane&15)][r][d0 + 2h + j]
//   B (4x16, KxN):  lane-half h, vgpr j -> W[r][d0 + 2j + h], broadcast over N
//   => every column of D holds the same 16 dot products; harvest N=0 (lane 0,
//      M=0..7) and N=0 upper half (lane 16, M=8..15).
//
// K split: 333 = 83*4 (unguarded main loop, max index 331) + 1 tail chunk at
// d0=332 where only k=0 is valid (zero-filled operands, branch-free selects).
// ---------------------------------------------------------------------------
__global__ void rout_kernel(const float* __restrict__ x,
                            const float* __restrict__ W,
                            const float* __restrict__ bias,
                            float* __restrict__ r_out)
{
    const int lane = threadIdx.x & 31;
    const int wid  = threadIdx.x >> 5;
    const int half = lane >> 4;      // 0: lanes 0-15, 1: lanes 16-31
    const int m    = lane & 15;      // A-matrix row (batch within tile)

    const size_t w  = (size_t)blockIdx.x * 8 + wid;  // global wave id
    const int    bt = (int)(w / D_R);                // batch tile
    const int    r  = (int)(w % D_R);                // region (fastest -> locality)
    const int    b0 = bt * 16;

    const float* __restrict__ xrow = x + ((size_t)(b0 + m) * D_R + r) * D_IN;
    const float* __restrict__ wrow = W + (size_t)r * D_IN;

    // Per-lane constant K offsets within each 4-chunk.
    const float* __restrict__ xa = xrow + 2 * half;  // A: k = 2h + j
    const float* __restrict__ wb = wrow + half;      // B: k = 2j + h

    v8f acc = {};

    // Main loop: 83 fully unguarded chunks (indices <= 331).
    for (int d0 = 0; d0 < 332; d0 += 4) {
        v2f a, bv;
        a[0]  = xa[d0];
        a[1]  = xa[d0 + 1];
        bv[0] = wb[d0];
        bv[1] = wb[d0 + 2];
        acc = __builtin_amdgcn_wmma_f32_16x16x4_f32(
            false, a, false, bv, (short)0, acc, false, false);
    }

    // Tail chunk d0=332: only k=0 (h=0, j=0) carries data; rest zero.
    {
        v2f a, bv;
        a[0]  = (half == 0) ? xrow[332] : 0.f;
        a[1]  = 0.f;
        bv[0] = (half == 0) ? wrow[332] : 0.f;
        bv[1] = 0.f;
        acc = __builtin_amdgcn_wmma_f32_16x16x4_f32(
            false, a, false, bv, (short)0, acc, false, false);
    }

    const float bb = bias[r];
    if (m == 0) {                                    // lanes 0 (M=0..7) and 16 (M=8..15)
        const size_t base = ((size_t)(b0 + half * 8)) * D_R + r;
        #pragma unroll
        for (int i = 0; i < 8; ++i)
            r_out[base + (size_t)i * D_R] = acc[i] + bb;
    }
}

// ---------------------------------------------------------------------------
// Kernel 3: out[b] = sum_r r_out[b,r] * r_att[r]   (one wave32 per batch)
// ---------------------------------------------------------------------------
__global__ void out_kernel(const float* __restrict__ r_out,
                           const float* __restrict__ att,
                           float* __restrict__ out)
{
    const int lane = threadIdx.x & 31;
    const int b    = blockIdx.x * (blockDim.x >> 5) + (threadIdx.x >> 5);
    const float* __restrict__ row = r_out + (size_t)b * D_R;

    float p = 0.f;
    for (int rr = lane; rr < D_R; rr += 32) p += row[rr] * att[rr];

    #pragma unroll
    for (int off = 16; off > 0; off >>= 1)
        p += __shfl_xor(p, off, 32);

    if (lane == 0) out[b] = p;
}

// ---------------------------------------------------------------------------
extern "C" void kernel_launch(void* const* d_in, const int* in_sizes, int n_in,
                              void* d_out, int out_size, void* d_ws, size_t ws_size,
                              hipStream_t stream)
{
    const float* x     = (const float*)d_in[0];  // [4096,333,333]
    const float* r_id  = (const float*)d_in[1];  // [333,333]
    const float* W     = (const float*)d_in[2];  // [333,333]
    const float* bias  = (const float*)d_in[3];  // [333]
    const float* att_w = (const float*)d_in[4];  // [333]

    float* out   = (float*)d_out;                         // [4096]
    float* r_out = out + BATCH;                           // [4096*333]
    float* r_att = r_out + (size_t)BATCH * D_R;           // [333]
    float* att_ws = (float*)d_ws;                         // 333 floats scratch

    att_kernel<<<1, 512, 0, stream>>>(r_id, att_w, att_ws, r_att);

    const int nwaves = (BATCH / 16) * D_R;                // 85,248
    rout_kernel<<<nwaves / 8, 256, 0, stream>>>(x, W, bias, r_out);

    out_kernel<<<BATCH / 8, 256, 0, stream>>>(r_out, att_ws, out);
}